// SRL_BERT_40183714021920
// MI455X (gfx1250) — compile-verified
//
#include <hip/hip_runtime.h>

typedef __attribute__((ext_vector_type(2))) float v2f;
typedef __attribute__((ext_vector_type(8))) float v8f;

#define B_ 16
#define S_ 512
#define H_ 768
#define R_ 8
#define D_ 512
#define C_ 27
#define W_ 256

// D = A(16x4 f32) * B(4x16 f32) + C(16x16 f32), wave32 striped.
// A layout: lanes 0-15 -> M=lane, {VGPR0:K=0, VGPR1:K=1}; lanes 16-31 -> K=2,3.
// B/C/D: row striped across lanes within a VGPR.
__device__ __forceinline__ v8f wmma_f32(v2f a, v2f b, v8f c) {
  return __builtin_amdgcn_wmma_f32_16x16x4_f32(
      /*neg_a=*/false, a, /*neg_b=*/false, b,
      /*c_mod=*/(short)0, c, /*reuse_a=*/false, /*reuse_b=*/false);
}

// ---------------- Kernel 1: rel_logits[b,s] = hs[b,s,:] . w_rel + b_rel ----
__global__ __launch_bounds__(256) void k_rel_logits(
    const float* __restrict__ hs, const float* __restrict__ w_rel,
    const float* __restrict__ b_rel, float* __restrict__ out) {
  int row  = blockIdx.x * 8 + (threadIdx.x >> 5);   // [0, B*S)
  int lane = threadIdx.x & 31;
  const float* h = hs + (size_t)row * H_;
  float acc = 0.f;
  for (int i = lane; i < H_; i += 32) acc += h[i] * w_rel[i];
  for (int off = 16; off > 0; off >>= 1) acc += __shfl_down(acc, off, 32);
  if (lane == 0) out[row] = acc + b_rel[0];
}

// ---------------- Kernel 2: zero first_hidden ------------------------------
__global__ __launch_bounds__(256) void k_zero4(float4* __restrict__ p, int n4) {
  for (int i = blockIdx.x * blockDim.x + threadIdx.x; i < n4;
       i += gridDim.x * blockDim.x)
    p[i] = make_float4(0.f, 0.f, 0.f, 0.f);
}

// ---------------- Kernel 3: scatter first-occurrence rows ------------------
// grid = B*S blocks, 192 threads (H/4 float4 per row). word_ids sorted per row.
__global__ __launch_bounds__(192) void k_scatter_first(
    const float* __restrict__ hs, const int* __restrict__ word_ids,
    float* __restrict__ FH) {
  int bs = blockIdx.x;
  int b = bs >> 9, s = bs & (S_ - 1);
  int wid = word_ids[bs];
  bool first = (s == 0) || (word_ids[bs - 1] != wid);
  if (!first) return;  // uniform across block
  const float4* src = (const float4*)(hs + (size_t)bs * H_);
  float4* dst = (float4*)(FH + ((size_t)(b * W_ + wid)) * H_);
  dst[threadIdx.x] = src[threadIdx.x];
}

// ---------------- Kernel 4: HW[m,n] = FH[m,:H] @ w1[:H,n] ------------------
// M = B*W = 4096, K = 768, N = 512. Block: 8 waves -> 128M x 64N tile.
// LDS B tile is K-pair interleaved: elem (kk,c) at ((kk>>1)*PS4 + c)*2 + (kk&1)
// so a B fragment (rows k,k+1 at col c) is one aligned ds_load_b64.
// 2*PS4 = 160 ≡ 32 (mod 64): the two lane-halves (pair-rows p, p+1) hit
// disjoint bank halves.
#define PS4 80
__global__ __launch_bounds__(256) void k_gemm_hword(
    const float* __restrict__ FH, const float* __restrict__ w1,
    float* __restrict__ HW) {
  __shared__ float ldsB[16 * PS4 * 2];  // 16 pair-rows (32 K) x 80 pairs
  const int tid = threadIdx.x;
  const int wave = tid >> 5, lane = tid & 31;
  const int mBase = blockIdx.x * 128 + wave * 16;
  const int n0 = blockIdx.y * 64;
  const int mRow = mBase + (lane & 15);
  const int klo = (lane >> 4) << 1;          // 0 or 2
  const float* aRow = FH + (size_t)mRow * H_;

  v8f acc[4];
  for (int j = 0; j < 4; ++j) acc[j] = (v8f)0.f;

  for (int k0 = 0; k0 < H_; k0 += 32) {
    for (int i = tid; i < 32 * 64; i += 256) {
      int kk = i >> 6, c = i & 63;
      ldsB[((kk >> 1) * PS4 + c) * 2 + (kk & 1)] =
          w1[(size_t)(k0 + kk) * D_ + n0 + c];
    }
    __syncthreads();
#pragma unroll
    for (int kk = 0; kk < 32; kk += 4) {
      v2f a = *(const v2f*)(aRow + k0 + kk + klo);
      const float* bp = &ldsB[(((kk + klo) >> 1) * PS4 + (lane & 15)) * 2];
#pragma unroll
      for (int j = 0; j < 4; ++j) {
        v2f bf = *(const v2f*)(bp + j * 32);   // one ds_load_b64
        acc[j] = wmma_f32(a, bf, acc[j]);
      }
    }
    __syncthreads();
  }

  const int rowOff = (lane >> 4) << 3;  // +8 for upper lane half
  const int colLane = lane & 15;
#pragma unroll
  for (int j = 0; j < 4; ++j) {
    int col = n0 + j * 16 + colLane;
#pragma unroll
    for (int g = 0; g < 8; ++g) {
      int row = mBase + g + rowOff;
      HW[(size_t)row * D_ + col] = acc[j][g];
    }
  }
}

// ---------------- Kernel 5: HR[b,r,:] = hs[b,pos,:] @ w1[H:,:] -------------
__global__ __launch_bounds__(256) void k_hrel(
    const float* __restrict__ hs, const int* __restrict__ relpos,
    const float* __restrict__ w1, float* __restrict__ HR) {
  __shared__ float sh[H_];
  int br = blockIdx.x;                    // b*R + r
  int b = br >> 3;
  int pos = relpos[br];
  const float* src = hs + ((size_t)b * S_ + pos) * H_;
  for (int i = threadIdx.x; i < H_; i += 256) sh[i] = src[i];
  __syncthreads();
  const float* w = w1 + (size_t)H_ * D_;  // second half of w1
  int d0 = threadIdx.x, d1 = threadIdx.x + 256;
  float a0 = 0.f, a1 = 0.f;
  for (int h = 0; h < H_; ++h) {
    float x = sh[h];
    a0 = fmaf(x, w[(size_t)h * D_ + d0], a0);
    a1 = fmaf(x, w[(size_t)h * D_ + d1], a1);
  }
  HR[(size_t)br * D_ + d0] = a0;
  HR[(size_t)br * D_ + d1] = a1;
}

// ---------------- Kernel 6: role_logits ------------------------------------
// Per (b,r): A = relu(HW[b,w,:] + HR[b,r,:] + b1) [256x512], B = w2 [512x27->32].
// Block: 8 waves cover 128 W-rows; grid = B*R*2.
// LDS w2 tile uses the same K-pair interleave; PS6=48 (2*48=96 ≡ 32 mod 64).
#define PS6 48
__global__ __launch_bounds__(256) void k_role(
    const float* __restrict__ HW, const float* __restrict__ HR,
    const float* __restrict__ b1, const float* __restrict__ w2,
    const float* __restrict__ b2, float* __restrict__ roleOut) {
  __shared__ float ldsW2[32 * PS6 * 2];  // 32 pair-rows (64 K) x 48 pairs
  __shared__ float ldsHR[D_];
  __shared__ float ldsB1[D_];

  const int tid = threadIdx.x;
  const int wave = tid >> 5, lane = tid & 31;
  const int br = blockIdx.x >> 1;         // b*R + r
  const int half = blockIdx.x & 1;
  const int b = br >> 3;
  const int mBase = half * 128 + wave * 16;    // w-row tile base
  const int klo = (lane >> 4) << 1;
  const int mRow = mBase + (lane & 15);
  const float* aRow = HW + (size_t)(b * W_ + mRow) * D_;

  for (int i = tid; i < D_; i += 256) {
    ldsHR[i] = HR[(size_t)br * D_ + i];
    ldsB1[i] = b1[i];
  }

  v8f acc[2];
  acc[0] = (v8f)0.f;
  acc[1] = (v8f)0.f;

  for (int k0 = 0; k0 < D_; k0 += 64) {
    __syncthreads();
    for (int i = tid; i < 64 * 32; i += 256) {
      int kk = i >> 5, c = i & 31;
      ldsW2[((kk >> 1) * PS6 + c) * 2 + (kk & 1)] =
          (c < C_) ? w2[(size_t)(k0 + kk) * C_ + c] : 0.f;
    }
    __syncthreads();
#pragma unroll
    for (int kk = 0; kk < 64; kk += 4) {
      int gk = k0 + kk + klo;                 // even -> 8B aligned LDS reads
      v2f hw = *(const v2f*)(aRow + gk);
      v2f hr = *(const v2f*)(&ldsHR[gk]);
      v2f bb = *(const v2f*)(&ldsB1[gk]);
      v2f a;
      a.x = fmaxf(hw.x + hr.x + bb.x, 0.f);
      a.y = fmaxf(hw.y + hr.y + bb.y, 0.f);
      const float* bp = &ldsW2[(((kk + klo) >> 1) * PS6 + (lane & 15)) * 2];
#pragma unroll
      for (int j = 0; j < 2; ++j) {
        v2f bf = *(const v2f*)(bp + j * 32);  // one ds_load_b64
        acc[j] = wmma_f32(a, bf, acc[j]);
      }
    }
  }

  const int rowOff = (lane >> 4) << 3;
  const int colLane = lane & 15;
#pragma unroll
  for (int j = 0; j < 2; ++j) {
    int col = j * 16 + colLane;
    if (col < C_) {
      float bias = b2[col];
#pragma unroll
      for (int g = 0; g < 8; ++g) {
        int row = mBase + g + rowOff;
        roleOut[((size_t)br * W_ + row) * C_ + col] = acc[j][g] + bias;
      }
    }
  }
}

extern "C" void kernel_launch(void* const* d_in, const int* in_sizes, int n_in,
                              void* d_out, int out_size, void* d_ws, size_t ws_size,
                              hipStream_t stream) {
  const float* hs     = (const float*)d_in[0];  // [B,S,H]
  const int*   wids   = (const int*)  d_in[1];  // [B,S]
  const int*   relpos = (const int*)  d_in[2];  // [B,R]
  const float* w_rel  = (const float*)d_in[3];  // [H]
  const float* b_rel  = (const float*)d_in[4];  // [1]
  const float* w1     = (const float*)d_in[5];  // [2H,D]
  const float* b1     = (const float*)d_in[6];  // [D]
  const float* w2     = (const float*)d_in[7];  // [D,C]
  const float* b2     = (const float*)d_in[8];  // [C]

  float* out_rel  = (float*)d_out;                       // [B,S]
  float* out_role = out_rel + (size_t)B_ * S_;           // [B,R,W,C]

  float* FH = (float*)d_ws;                              // [B,W,H]
  float* HW = FH + (size_t)B_ * W_ * H_;                 // [B,W,D]
  float* HR = HW + (size_t)B_ * W_ * D_;                 // [B,R,D]

  // 1) relational logits
  k_rel_logits<<<(B_ * S_) / 8, 256, 0, stream>>>(hs, w_rel, b_rel, out_rel);

  // 2) zero first_hidden
  int n4 = (B_ * W_ * H_) / 4;
  k_zero4<<<1024, 256, 0, stream>>>((float4*)FH, n4);

  // 3) scatter first-occurrence token embeddings
  k_scatter_first<<<B_ * S_, H_ / 4, 0, stream>>>(hs, wids, FH);

  // 4) h_word GEMM (WMMA f32): [4096 x 768] @ [768 x 512]
  k_gemm_hword<<<dim3((B_ * W_) / 128, D_ / 64), 256, 0, stream>>>(FH, w1, HW);

  // 5) relation-token projection: [B*R x 768] @ [768 x 512]
  k_hrel<<<B_ * R_, 256, 0, stream>>>(hs, relpos, w1, HR);

  // 6) role logits (WMMA f32 with fused relu epilogue input)
  k_role<<<B_ * R_ * 2, 256, 0, stream>>>(HW, HR, b1, w2, b2, out_role);
}